// SplitLinear_33732673143349
// MI455X (gfx1250) — compile-verified
//
#include <hip/hip_runtime.h>
#include <hip/hip_bf16.h>

// ---------------------------------------------------------------------------
// SplitLinear collapsed form:
//   S = X @ [Wfc ; Wsel]^T            (M=4096, N=420->448 pad, K=4096)  [WMMA]
//   c[f,n] = Wfc[f,:] . Wexp[:,n]     (410x10, K=4096)
//   d[f]   = Wfc[f,:] . bexp + bfc[f]
//   out[m, n*410+f] = S[m,f] + (S[m,410+n]+bsel[n])*c[f,n] + d[f],  j<4096
// ---------------------------------------------------------------------------

#define IN_F   4096
#define M_TOK  4096          // 2*2048 tokens
#define SPLT   410
#define NPART  10
#define NREAL  420           // 410 fc cols + 10 selector cols
#define NPAD   448           // padded to 28 * 16 WMMA tiles
#define OUTF   4096

typedef __attribute__((ext_vector_type(2))) float v2f;
typedef __attribute__((ext_vector_type(8))) float v8f;

// --- kernel 1: pack [Wfc ; Wsel ; zeros] into Wpack[NPAD][IN_F] --------------
__global__ __launch_bounds__(256) void pack_kernel(const float* __restrict__ Wfc,
                                                   const float* __restrict__ Wsel,
                                                   float* __restrict__ Wp) {
    int i   = blockIdx.x * blockDim.x + threadIdx.x;   // float4 index
    int row = i >> 10;                                 // IN_F/4 = 1024 float4 per row
    int c4  = i & 1023;
    if (row >= NPAD) return;
    float4 v;
    if (row < SPLT)        v = ((const float4*)(Wfc  + (size_t)row * IN_F))[c4];
    else if (row < NREAL)  v = ((const float4*)(Wsel + (size_t)(row - SPLT) * IN_F))[c4];
    else                   v = make_float4(0.f, 0.f, 0.f, 0.f);
    ((float4*)(Wp + (size_t)row * IN_F))[c4] = v;
}

// --- kernel 2: c[f][n] and d[f] (one workgroup per f) ------------------------
__global__ __launch_bounds__(256) void cd_kernel(const float* __restrict__ Wfc,
                                                 const float* __restrict__ Wexp,
                                                 const float* __restrict__ bexp,
                                                 const float* __restrict__ bfc,
                                                 float* __restrict__ cmat,
                                                 float* __restrict__ dvec) {
    __shared__ float red[256];
    const int f = blockIdx.x;
    const int t = threadIdx.x;
    const float* wrow = Wfc + (size_t)f * IN_F;
    float acc[NPART + 1];
#pragma unroll
    for (int o = 0; o <= NPART; ++o) acc[o] = 0.f;
    for (int k = t; k < IN_F; k += 256) {
        float w = wrow[k];
        const float* we = Wexp + (size_t)k * NPART;
#pragma unroll
        for (int n = 0; n < NPART; ++n) acc[n] += w * we[n];
        acc[NPART] += w * bexp[k];
    }
    for (int o = 0; o <= NPART; ++o) {
        red[t] = acc[o];
        __syncthreads();
        for (int s = 128; s > 0; s >>= 1) {
            if (t < s) red[t] += red[t + s];
            __syncthreads();
        }
        if (t == 0) {
            if (o < NPART) cmat[f * NPART + o] = red[0];
            else           dvec[f] = red[0] + bfc[f];
        }
        __syncthreads();
    }
}

// --- kernel 3: WMMA f32 GEMM  S[M_TOK][NPAD] = X @ Wp^T ----------------------
// Wave tile 32x32 (2x2 of 16x16), 4 waves per WG -> 64x64 per WG.
// A 16x4 frag: lane<16 holds {K=0,K=1}, lane>=16 holds {K=2,K=3} (ISA layout).
// B 4x16 frag mirrors it with N striped across lanes, so B[k][n] = Wp[n][k]
// loads identically to A from the row-major packed weights.
__global__ __launch_bounds__(128) void gemm_wmma(const float* __restrict__ X,
                                                 const float* __restrict__ Wp,
                                                 float* __restrict__ S) {
    const int lane  = threadIdx.x & 31;
    const int wave  = threadIdx.x >> 5;
    const int l16   = lane & 15;
    const int khalf = (lane >> 4) << 1;                  // 0 or 2
    const int mBase = blockIdx.x * 64 + (wave >> 1) * 32;
    const int nBase = blockIdx.y * 64 + (wave &  1) * 32;

    const float* a0p = X  + (size_t)(mBase      + l16) * IN_F + khalf;
    const float* a1p = X  + (size_t)(mBase + 16 + l16) * IN_F + khalf;
    const float* b0p = Wp + (size_t)(nBase      + l16) * IN_F + khalf;
    const float* b1p = Wp + (size_t)(nBase + 16 + l16) * IN_F + khalf;

    v8f acc00 = {}, acc01 = {}, acc10 = {}, acc11 = {};

#pragma unroll 8
    for (int k = 0; k < IN_F; k += 4) {
        v2f a0 = *(const v2f*)(a0p + k);
        v2f a1 = *(const v2f*)(a1p + k);
        v2f b0 = *(const v2f*)(b0p + k);
        v2f b1 = *(const v2f*)(b1p + k);
        acc00 = __builtin_amdgcn_wmma_f32_16x16x4_f32(false, a0, false, b0, (short)0, acc00, false, false);
        acc01 = __builtin_amdgcn_wmma_f32_16x16x4_f32(false, a0, false, b1, (short)0, acc01, false, false);
        acc10 = __builtin_amdgcn_wmma_f32_16x16x4_f32(false, a1, false, b0, (short)0, acc10, false, false);
        acc11 = __builtin_amdgcn_wmma_f32_16x16x4_f32(false, a1, false, b1, (short)0, acc11, false, false);
    }

    const int rowOff = (lane >> 4) << 3;                 // lanes>=16 hold M = r+8
    auto store_tile = [&](const v8f& c, int mt, int nt) {
        float* p = S + (size_t)(mBase + mt * 16 + rowOff) * NPAD + (nBase + nt * 16 + l16);
#pragma unroll
        for (int r = 0; r < 8; ++r) p[(size_t)r * NPAD] = c[r];
    };
    store_tile(acc00, 0, 0);
    store_tile(acc01, 0, 1);
    store_tile(acc10, 1, 0);
    store_tile(acc11, 1, 1);
}

// --- kernel 4: epilogue, coalesced stream-out of 64 MiB ----------------------
__global__ __launch_bounds__(256) void epi_kernel(const float* __restrict__ S,
                                                  const float* __restrict__ cmat,
                                                  const float* __restrict__ dvec,
                                                  const float* __restrict__ bsel,
                                                  float* __restrict__ out) {
    int idx = blockIdx.x * blockDim.x + threadIdx.x;     // [0, M_TOK*OUTF)
    int m = idx >> 12;                                   // OUTF = 4096
    int j = idx & (OUTF - 1);
    int n = j / SPLT;                                    // 0..9
    int f = j - n * SPLT;                                // 0..409
    const float* srow = S + (size_t)m * NPAD;
    float sx = srow[SPLT + n] + bsel[n];
    out[idx] = srow[f] + sx * cmat[f * NPART + n] + dvec[f];
}

extern "C" void kernel_launch(void* const* d_in, const int* in_sizes, int n_in,
                              void* d_out, int out_size, void* d_ws, size_t ws_size,
                              hipStream_t stream) {
    const float* x    = (const float*)d_in[0];
    const float* Wsel = (const float*)d_in[1];
    const float* bsel = (const float*)d_in[2];
    const float* Wexp = (const float*)d_in[3];
    const float* bexp = (const float*)d_in[4];
    const float* Wfc  = (const float*)d_in[5];
    const float* bfc  = (const float*)d_in[6];
    float* out = (float*)d_out;

    // workspace layout (bytes)
    char* ws = (char*)d_ws;
    float* Wp   = (float*)(ws);                                         // NPAD * IN_F
    float* S    = (float*)(ws + (size_t)NPAD * IN_F * sizeof(float));   // M_TOK * NPAD
    float* cmat = (float*)(ws + (size_t)(NPAD * IN_F + (size_t)M_TOK * NPAD) * sizeof(float));
    float* dvec = cmat + SPLT * NPART;

    // 1. pack weights: [Wfc ; Wsel ; 0]
    {
        int n4 = NPAD * (IN_F / 4);
        pack_kernel<<<(n4 + 255) / 256, 256, 0, stream>>>(Wfc, Wsel, Wp);
    }
    // 2. c[f,n], d[f]
    cd_kernel<<<SPLT, 256, 0, stream>>>(Wfc, Wexp, bexp, bfc, cmat, dvec);
    // 3. S = X @ Wp^T via v_wmma_f32_16x16x4_f32
    {
        dim3 grid(M_TOK / 64, NPAD / 64);   // 64 x 7
        gemm_wmma<<<grid, 128, 0, stream>>>(x, Wp, S);
    }
    // 4. epilogue
    {
        int total = M_TOK * OUTF;
        epi_kernel<<<total / 256, 256, 0, stream>>>(S, cmat, dvec, bsel, out);
    }
}